// SaConv_21045339750971
// MI455X (gfx1250) — compile-verified
//
#include <hip/hip_runtime.h>
#include <hip/hip_bf16.h>

typedef __attribute__((ext_vector_type(4))) float vf4;
typedef __attribute__((ext_vector_type(2))) float v2f;
typedef __attribute__((ext_vector_type(8))) float v8f;

#define N_NODES 100000
#define DIM     128
#define QKD     32
#define N_EDGES 1600000
#define KSLOTS  8          // a_feat + 7 hops
#define SCALE   0.08838834764831845f   // 1/sqrt(128)

// Native fp32 global atomic add (no CAS loop, no return -> STOREcnt only).
__device__ __forceinline__ void atomic_add_f32(float* p, float v) {
    unsafeAtomicAdd(p, v);
}

// ---------------- utility ----------------
__global__ void k_zero(float* __restrict__ p, long n4) {
    long i = (long)blockIdx.x * blockDim.x + threadIdx.x;
    long stride = (long)gridDim.x * blockDim.x;
    vf4 z = {0.f, 0.f, 0.f, 0.f};
    for (; i < n4; i += stride) ((vf4*)p)[i] = z;
}

// deg[row[e]] += 1
__global__ void k_degree(const int* __restrict__ row, float* __restrict__ deg, int E) {
    int e = blockIdx.x * blockDim.x + threadIdx.x;
    if (e < E) atomic_add_f32(&deg[row[e]], 1.0f);
}

// deg -> rsqrt(max(deg,1)) in place
__global__ void k_dinv(float* __restrict__ deg, int N) {
    int n = blockIdx.x * blockDim.x + threadIdx.x;
    if (n < N) {
        float d = deg[n];
        d = d < 1.0f ? 1.0f : d;
        deg[n] = rsqrtf(d);
    }
}

// ---------------- propagation ----------------
// one wave (32 lanes) per edge; lane handles 4 consecutive feats
__global__ void k_scatter(const float* __restrict__ x, const int* __restrict__ row,
                          const int* __restrict__ col, const float* __restrict__ dinv,
                          float* __restrict__ acc, int E) {
    long t = (long)blockIdx.x * blockDim.x + threadIdx.x;
    int e = (int)(t >> 5);
    int lane = (int)(t & 31);
    if (e >= E) return;
    int r = row[e];
    int c = col[e];
    float s = dinv[r];
    vf4 v = *(const vf4*)(x + (size_t)r * DIM + lane * 4);
    float* dst = acc + (size_t)c * DIM + lane * 4;
    atomic_add_f32(dst + 0, v[0] * s);
    atomic_add_f32(dst + 1, v[1] * s);
    atomic_add_f32(dst + 2, v[2] * s);
    atomic_add_f32(dst + 3, v[3] * s);
}

// pass 0: t = dinv*acc (== P x0); L0 = x0 + t; x1 = x0 - t; L1 = x1
__global__ void k_combine0(const float* __restrict__ x0, const float* __restrict__ acc,
                           const float* __restrict__ dinv, float* __restrict__ L,
                           float* __restrict__ x, int N) {
    long t = (long)blockIdx.x * blockDim.x + threadIdx.x;
    int n = (int)(t >> 5);
    int lane = (int)(t & 31);
    if (n >= N) return;
    float s = dinv[n];
    size_t fo = (size_t)n * DIM + lane * 4;
    vf4 t4 = *(const vf4*)(acc + fo);
    t4 *= s;
    vf4 x4 = *(const vf4*)(x0 + fo);
    vf4 l0 = x4 + t4;
    vf4 x1 = x4 - t4;
    size_t lo = (size_t)n * (KSLOTS * DIM) + lane * 4;
    __builtin_nontemporal_store(l0, (vf4*)(L + lo));            // slot 0
    __builtin_nontemporal_store(x1, (vf4*)(L + lo + DIM));      // slot 1
    *(vf4*)(x + fo) = x1;
}

// hop pass: xn = x - dinv*acc; L[slot] = xn; x = xn
__global__ void k_combine(const float* __restrict__ acc, const float* __restrict__ dinv,
                          float* __restrict__ L, float* __restrict__ x, int slot, int N) {
    long t = (long)blockIdx.x * blockDim.x + threadIdx.x;
    int n = (int)(t >> 5);
    int lane = (int)(t & 31);
    if (n >= N) return;
    float s = dinv[n];
    size_t fo = (size_t)n * DIM + lane * 4;
    vf4 t4 = *(const vf4*)(acc + fo);
    vf4 x4 = *(const vf4*)(x + fo);
    vf4 xn = x4 - t4 * s;
    *(vf4*)(x + fo) = xn;
    __builtin_nontemporal_store(xn, (vf4*)(L + (size_t)n * (KSLOTS * DIM) + (size_t)slot * DIM + lane * 4));
}

// ---------------- query prep ----------------
// per-column partial sums of h_ori, atomically folded into hbar[128]
__global__ void k_rowsum(const float* __restrict__ h, float* __restrict__ hbar, int N) {
    int d = threadIdx.x;            // 128 threads
    int n0 = blockIdx.x * 256;
    float local = 0.f;
    for (int i = 0; i < 256; ++i) {
        int n = n0 + i;
        if (n < N) local += h[(size_t)n * DIM + d];
    }
    atomic_add_f32(&hbar[d], local);
}

// single block, 128 threads: qg = hbar/N @ Wq + bq ; wkq = Wk @ qg ; c = bk.qg
__global__ void k_qprep(const float* __restrict__ hbar, const float* __restrict__ Wq,
                        const float* __restrict__ bq, const float* __restrict__ Wk,
                        const float* __restrict__ bk, float* __restrict__ qg,
                        float* __restrict__ wkq, float* __restrict__ cterm, float invN) {
    __shared__ float sq[QKD];
    int t = threadIdx.x;
    if (t < QKD) {
        float acc = 0.f;
        for (int d = 0; d < DIM; ++d) acc += hbar[d] * Wq[d * QKD + t];
        float q = acc * invN + bq[t];
        sq[t] = q;
        qg[t] = q;
    }
    __syncthreads();
    float acc = 0.f;
    for (int k = 0; k < QKD; ++k) acc += Wk[t * QKD + k] * sq[k];
    wkq[t] = acc;
    if (t == 0) {
        float cc = 0.f;
        for (int k = 0; k < QKD; ++k) cc += bk[k] * sq[k];
        *cterm = cc;
    }
}

// ---------------- attention (softmax over 8 hops), y = sum_k A_k * L_k ----------------
__global__ void k_attn(const float* __restrict__ L, const float* __restrict__ wkq,
                       const float* __restrict__ cterm, float* __restrict__ y, int N) {
    long t = (long)blockIdx.x * blockDim.x + threadIdx.x;
    int n = (int)(t >> 5);
    int lane = (int)(t & 31);
    if (n >= N) return;
    vf4 w = *(const vf4*)(wkq + lane * 4);
    const float* base = L + (size_t)n * (KSLOTS * DIM) + lane * 4;
    vf4 lv[KSLOTS];
    float s[KSLOTS];
#pragma unroll
    for (int k = 0; k < KSLOTS; ++k) {
        lv[k] = __builtin_nontemporal_load((const vf4*)(base + (size_t)k * DIM));
        s[k] = lv[k][0] * w[0] + lv[k][1] * w[1] + lv[k][2] * w[2] + lv[k][3] * w[3];
    }
    // butterfly reduce each of the 8 dots across the wave32
#pragma unroll
    for (int m = 16; m > 0; m >>= 1) {
#pragma unroll
        for (int k = 0; k < KSLOTS; ++k) s[k] += __shfl_xor(s[k], m, 32);
    }
    float c = *cterm;
    float mx = -1e30f;
#pragma unroll
    for (int k = 0; k < KSLOTS; ++k) {
        s[k] = SCALE * (s[k] + c);
        mx = s[k] > mx ? s[k] : mx;
    }
    float sum = 0.f;
#pragma unroll
    for (int k = 0; k < KSLOTS; ++k) {
        s[k] = __expf(s[k] - mx);
        sum += s[k];
    }
    float inv = 1.0f / sum;
    vf4 acc = {0.f, 0.f, 0.f, 0.f};
#pragma unroll
    for (int k = 0; k < KSLOTS; ++k) acc += lv[k] * (s[k] * inv);
    *(vf4*)(y + (size_t)n * DIM + lane * 4) = acc;
}

// ---------------- final GEMM: out = y[N,128] @ Wv[128,128] + bv, via WMMA f32 16x16x4 ----------------
// One wave per 16-row strip; 8 column tiles of 16; 32 K-steps of 4.
__global__ void k_gemm_wmma(const float* __restrict__ y, const float* __restrict__ Wv,
                            const float* __restrict__ bv, float* __restrict__ out, int N) {
    long t = (long)blockIdx.x * blockDim.x + threadIdx.x;
    int wave = (int)(t >> 5);
    int lane = threadIdx.x & 31;
    int row0 = wave * 16;
    if (row0 >= N) return;                 // wave-uniform: EXEC stays all-ones below
    int mlane = lane & 15;
    int khalf = lane >> 4;                 // 0: K+{0,1}, 1: K+{2,3}  (ISA A 16x4 layout)
    int r = row0 + mlane;
    if (r > N - 1) r = N - 1;              // clamp reads (uniform control flow)
    const float* yrow = y + (size_t)r * DIM;

    v2f a[32];
#pragma unroll
    for (int kk = 0; kk < 32; ++kk)
        a[kk] = *(const v2f*)(yrow + kk * 4 + khalf * 2);

    for (int ct = 0; ct < 8; ++ct) {
        int col = ct * 16 + mlane;
        float bvv = bv[col];
        v8f c;
#pragma unroll
        for (int i = 0; i < 8; ++i) c[i] = bvv;   // bias broadcast over rows
#pragma unroll
        for (int kk = 0; kk < 32; ++kk) {
            int kb = kk * 4 + khalf * 2;          // ISA B 4x16 layout: VGPR0=K{0,2}, VGPR1=K{1,3}
            v2f b;
            b[0] = Wv[(size_t)kb * DIM + col];
            b[1] = Wv[(size_t)(kb + 1) * DIM + col];
            c = __builtin_amdgcn_wmma_f32_16x16x4_f32(false, a[kk], false, b,
                                                      (short)0, c, false, false);
        }
        // D layout: VGPR i -> M = i + khalf*8, N = mlane
#pragma unroll
        for (int i = 0; i < 8; ++i) {
            int rr = row0 + khalf * 8 + i;
            if (rr < N) out[(size_t)rr * DIM + ct * 16 + mlane] = c[i];
        }
    }
}

// ---------------- host ----------------
extern "C" void kernel_launch(void* const* d_in, const int* in_sizes, int n_in,
                              void* d_out, int out_size, void* d_ws, size_t ws_size,
                              hipStream_t stream) {
    const float* h_train = (const float*)d_in[0];
    const float* h_ori   = (const float*)d_in[1];
    const int*   edges   = (const int*)d_in[2];     // [2,E] row-major: row then col
    const float* Wq      = (const float*)d_in[3];
    const float* bq      = (const float*)d_in[4];
    const float* Wk      = (const float*)d_in[5];
    const float* bk      = (const float*)d_in[6];
    const float* Wv      = (const float*)d_in[7];
    const float* bv      = (const float*)d_in[8];
    float* out = (float*)d_out;

    const int N = N_NODES, E = N_EDGES;
    const int* row = edges;
    const int* col = edges + E;

    // workspace carve-up (256B aligned)
    char* ws = (char*)d_ws;
    size_t off = 0;
    auto carve = [&](size_t bytes) -> float* {
        float* p = (float*)(ws + off);
        off = (off + bytes + 255) & ~(size_t)255;
        return p;
    };
    float* dinv  = carve((size_t)N * sizeof(float));                  // deg -> dinv
    float* acc   = carve((size_t)N * DIM * sizeof(float));            // propagate accumulator
    float* xbuf  = carve((size_t)N * DIM * sizeof(float));            // current x; reused as y
    float* stats = carve(512 * sizeof(float));                        // hbar(128)+qg(32)+wkq(128)+c(1)
    float* Lst   = carve((size_t)N * KSLOTS * DIM * sizeof(float));   // [N,8,128] streamed (NT)
    float* hbar = stats;
    float* qg   = stats + 128;
    float* wkq  = stats + 160;
    float* ct   = stats + 288;
    float* ybuf = xbuf;   // x no longer needed once L is full

    const int TB = 256;
    dim3 blk(TB);
    int zgrid = 1024;

    // degrees + stats zero
    k_zero<<<zgrid, blk, 0, stream>>>(dinv, (long)N / 4);
    k_zero<<<1, 128, 0, stream>>>(stats, 512 / 4);
    k_degree<<<(E + TB - 1) / TB, blk, 0, stream>>>(row, dinv, E);
    k_dinv<<<(N + TB - 1) / TB, blk, 0, stream>>>(dinv, N);

    // global-mean query prep (independent of propagation)
    k_rowsum<<<(N + 255) / 256, dim3(128), 0, stream>>>(h_ori, hbar, N);
    k_qprep<<<1, dim3(128), 0, stream>>>(hbar, Wq, bq, Wk, bk, qg, wkq, ct, 1.0f / (float)N);

    long nwork = (long)N * 32;
    int ngrid = (int)((nwork + TB - 1) / TB);
    long ework = (long)E * 32;
    int egrid = (int)((ework + TB - 1) / TB);

    // pass 0: P x0 -> L0 = x0 + Px0, x1 = x0 - Px0 (stored as L1 too)
    k_zero<<<zgrid, blk, 0, stream>>>(acc, (long)N * DIM / 4);
    k_scatter<<<egrid, blk, 0, stream>>>(h_train, row, col, dinv, acc, E);
    k_combine0<<<ngrid, blk, 0, stream>>>(h_train, acc, dinv, Lst, xbuf, N);

    // hops 2..7
    for (int slot = 2; slot < KSLOTS; ++slot) {
        k_zero<<<zgrid, blk, 0, stream>>>(acc, (long)N * DIM / 4);
        k_scatter<<<egrid, blk, 0, stream>>>(xbuf, row, col, dinv, acc, E);
        k_combine<<<ngrid, blk, 0, stream>>>(acc, dinv, Lst, xbuf, slot, N);
    }

    // softmax attention over hops -> y
    k_attn<<<ngrid, blk, 0, stream>>>(Lst, wkq, ct, ybuf, N);

    // out = y @ Wv + bv via WMMA
    int waves = (N + 15) / 16;
    long gwork = (long)waves * 32;
    int ggrid = (int)((gwork + TB - 1) / TB);
    k_gemm_wmma<<<ggrid, blk, 0, stream>>>(ybuf, Wv, bv, out, N);
}